// CompGCN_30846455119993
// MI455X (gfx1250) — compile-verified
//
#include <hip/hip_runtime.h>
#include <hip/hip_bf16.h>

// CompGCN, restructured:
//   agg = segsum_dir0(h[src]-h_r[et]) @ W_O^T + segsum_dir1(h[src]-h_r[et]) @ W_I^T
// Edge pass is pure gather + f32 atomic scatter (L2 resident on MI455X: 192MB L2
// holds h (25.6MB) + acc (51.2MB) + weights).  Node pass is dense [N,64]x[64,64]
// GEMMs done with V_WMMA_F32_16X16X4_F32 (fp32-exact, compute is negligible).

typedef __attribute__((ext_vector_type(2))) float v2f;
typedef __attribute__((ext_vector_type(8))) float v8f;

#define D 64

// ---------------------------------------------------------------- h_r = alpha @ Basis
__global__ void hr_init_kernel(const float* __restrict__ Basis,   // [16,64]
                               const float* __restrict__ alpha,   // [3,16]
                               float* __restrict__ h_r)           // [3,64]
{
    int t = threadIdx.x;              // 0..191
    if (t >= 3 * D) return;
    int r = t >> 6, n = t & 63;
    float s = 0.f;
#pragma unroll
    for (int b = 0; b < 16; ++b) s += alpha[r * 16 + b] * Basis[b * D + n];
    h_r[t] = s;
}

// ---------------------------------------------------------------- h_r' = act(h_r @ Wrel^T)
__global__ void rel_kernel(const float* __restrict__ h_r,    // [3,64]
                           const float* __restrict__ Wrel,   // [64,64]
                           float* __restrict__ h_r_out,      // [3,64]
                           int relu)
{
    int t = threadIdx.x;
    if (t >= 3 * D) return;
    int r = t >> 6, n = t & 63;
    float s = 0.f;
#pragma unroll 8
    for (int k = 0; k < D; ++k) s += h_r[r * D + k] * Wrel[n * D + k];
    if (relu) s = fmaxf(s, 0.f);
    h_r_out[t] = s;
}

// ---------------------------------------------------------------- zero accumulators
__global__ __launch_bounds__(256) void zero_kernel(float4* __restrict__ p, long long n4)
{
    long long i = (long long)blockIdx.x * blockDim.x + threadIdx.x;
    if (i < n4) p[i] = make_float4(0.f, 0.f, 0.f, 0.f);
}

// ---------------------------------------------------------------- edge scatter pass
// One thread per (edge, 4 channels): b128 gather of h[src], LDS-resident h_r
// subtract, 4x global_atomic_add_f32 into acc[dir][dst].  16 threads/edge, so
// index loads are shared 4x better than channel-per-thread and the gather is a
// single global_load_b128.
__global__ __launch_bounds__(256)
void edge_kernel(const float* __restrict__ h_in,
                 const float* __restrict__ h_r,      // [3,64] (rows 0..1 used)
                 const int* __restrict__ src,
                 const int* __restrict__ dst,
                 const int* __restrict__ etype,
                 const int* __restrict__ dir,
                 float* __restrict__ acc,            // [2, n_nodes, 64]
                 long long total16, int n_nodes)
{
    __shared__ float lds_hr[2 * D];
    int t = threadIdx.x;
    if (t < 2 * D) lds_hr[t] = h_r[t];
    __syncthreads();

    long long i = (long long)blockIdx.x * blockDim.x + t;
    if (i >= total16) return;
    int e = (int)(i >> 4);
    int c = (int)(i & 15) * 4;
    int s  = src[e];
    int d0 = dst[e];
    int et = etype[e];
    int dr = dir[e];

    const float4 hv = *(const float4*)(h_in + (size_t)s * D + c);
    const float4 rv = *(const float4*)(lds_hr + et * D + c);

    float* p = acc + ((size_t)dr * n_nodes + (size_t)d0) * D + c;
    atomicAdd(p + 0, hv.x - rv.x);
    atomicAdd(p + 1, hv.y - rv.y);
    atomicAdd(p + 2, hv.z - rv.z);
    atomicAdd(p + 3, hv.w - rv.w);
}

// ---------------------------------------------------------------- node pass (WMMA)
// h_out = act( (h_in - hr2) @ Ws^T + acc_out @ Wo^T + acc_in @ Wi^T )
// Block: 256 threads = 8 waves.  Block covers 32 nodes; wave w handles node tile
// (w>>2) and out-channel tile (w&3)*16.  Weights staged in LDS (48 KB).
__global__ __launch_bounds__(256)
void node_kernel(const float* __restrict__ h_in,
                 const float* __restrict__ acc,     // [2, n_nodes, 64]
                 const float* __restrict__ hr2,     // 64 floats (self-loop relation)
                 const float* __restrict__ Ws,
                 const float* __restrict__ Wo,
                 const float* __restrict__ Wi,
                 float* __restrict__ h_out,
                 int n_nodes, int relu)
{
    __shared__ float lds_w[3 * D * D];   // 48 KB
    __shared__ float lds_hr2[D];

    int t = threadIdx.x;
    for (int i = t; i < D * D; i += 256) {
        lds_w[i]             = Ws[i];
        lds_w[D * D + i]     = Wo[i];
        lds_w[2 * D * D + i] = Wi[i];
    }
    if (t < D) lds_hr2[t] = hr2[t];
    __syncthreads();

    const int wave  = t >> 5;
    const int lane  = t & 31;
    const int lhalf = lane >> 4;   // 0 | 1
    const int lmod  = lane & 15;

    const int m0 = (blockIdx.x * 2 + (wave >> 2)) * 16;  // 16-node tile base
    const int n0 = (wave & 3) * 16;                      // 16-outchan tile base
    if (m0 >= n_nodes) return;

    const int arow = m0 + lmod;                          // A-matrix row of this lane
    const int brow = n0 + lmod;                          // weight row (out channel)

    const float* __restrict__ xr = h_in + (size_t)arow * D;
    const float* __restrict__ a0 = acc + (size_t)arow * D;
    const float* __restrict__ a1 = acc + ((size_t)n_nodes + (size_t)arow) * D;

    float res[8];

#if defined(__has_builtin) && __has_builtin(__builtin_amdgcn_wmma_f32_16x16x4_f32)
    v8f c = {};
    // A 16x4 layout: lanes 0-15 -> K={k0,k0+1}, lanes 16-31 -> K={k0+2,k0+3}, row = lane%16
    // B 4x16 layout mirrored: B[k][n] = W[n0 + lane%16][k]
#pragma unroll
    for (int k0 = 0; k0 < D; k0 += 4) {
        const int ka = k0 + lhalf * 2;
        v2f a, b;
        a.x = xr[ka]     - lds_hr2[ka];
        a.y = xr[ka + 1] - lds_hr2[ka + 1];
        b.x = lds_w[brow * D + ka];
        b.y = lds_w[brow * D + ka + 1];
        c = __builtin_amdgcn_wmma_f32_16x16x4_f32(false, a, false, b, (short)0, c, false, false);
    }
#pragma unroll
    for (int k0 = 0; k0 < D; k0 += 4) {
        const int ka = k0 + lhalf * 2;
        v2f a, b;
        a.x = a0[ka];
        a.y = a0[ka + 1];
        b.x = lds_w[D * D + brow * D + ka];
        b.y = lds_w[D * D + brow * D + ka + 1];
        c = __builtin_amdgcn_wmma_f32_16x16x4_f32(false, a, false, b, (short)0, c, false, false);
    }
#pragma unroll
    for (int k0 = 0; k0 < D; k0 += 4) {
        const int ka = k0 + lhalf * 2;
        v2f a, b;
        a.x = a1[ka];
        a.y = a1[ka + 1];
        b.x = lds_w[2 * D * D + brow * D + ka];
        b.y = lds_w[2 * D * D + brow * D + ka + 1];
        c = __builtin_amdgcn_wmma_f32_16x16x4_f32(false, a, false, b, (short)0, c, false, false);
    }
#pragma unroll
    for (int r = 0; r < 8; ++r) res[r] = c[r];
#else
    // scalar fallback (keeps compiling if the builtin is absent)
#pragma unroll
    for (int r = 0; r < 8; ++r) {
        const int m = m0 + r + lhalf * 8;
        const float* __restrict__ xm = h_in + (size_t)m * D;
        const float* __restrict__ b0 = acc + (size_t)m * D;
        const float* __restrict__ b1 = acc + ((size_t)n_nodes + (size_t)m) * D;
        float s = 0.f;
        for (int k = 0; k < D; ++k) {
            s += (xm[k] - lds_hr2[k]) * lds_w[brow * D + k];
            s += b0[k] * lds_w[D * D + brow * D + k];
            s += b1[k] * lds_w[2 * D * D + brow * D + k];
        }
        res[r] = s;
    }
#endif

    // C/D layout: VGPR r, M = m0 + r + 8*lhalf, N = n0 + lane%16
#pragma unroll
    for (int r = 0; r < 8; ++r) {
        float v = res[r];
        if (relu) v = fmaxf(v, 0.f);
        const int m = m0 + r + lhalf * 8;
        h_out[(size_t)m * D + n0 + lmod] = v;
    }
}

// ---------------------------------------------------------------- launch
extern "C" void kernel_launch(void* const* d_in, const int* in_sizes, int n_in,
                              void* d_out, int out_size, void* d_ws, size_t ws_size,
                              hipStream_t stream)
{
    const float* h_u   = (const float*)d_in[0];
    const float* Basis = (const float*)d_in[1];
    const float* alpha = (const float*)d_in[2];
    const float* W_O   = (const float*)d_in[3];   // [3,64,64]
    const float* W_I   = (const float*)d_in[4];
    const float* W_S   = (const float*)d_in[5];
    const float* W_rel = (const float*)d_in[6];
    const int*   src   = (const int*)d_in[7];
    const int*   dst   = (const int*)d_in[8];
    const int*   etype = (const int*)d_in[9];
    const int*   edir  = (const int*)d_in[10];
    float* out = (float*)d_out;

    const int n_nodes = in_sizes[0] / D;     // 100000
    const int n_edges = in_sizes[7];         // 1200000

    // workspace layout
    float* acc = (float*)d_ws;                               // [2, N, 64]  51.2 MB
    float* hb0 = acc + (size_t)2 * n_nodes * D;              // [N, 64]     25.6 MB
    float* hb1 = hb0 + (size_t)n_nodes * D;                  // [N, 64]     25.6 MB
    float* hra = hb1 + (size_t)n_nodes * D;                  // [3, 64]
    float* hrb = hra + 256;

    hr_init_kernel<<<1, 192, 0, stream>>>(Basis, alpha, hra);

    const float* hin[3]  = { h_u, hb0, hb1 };
    float*       hout[3] = { hb0, hb1, out };

    const long long edge_total16 = (long long)n_edges * 16;   // threads: 4 ch each
    const unsigned  edge_blocks  = (unsigned)((edge_total16 + 255) / 256);
    const long long acc_f4       = (long long)2 * n_nodes * D / 4;
    const unsigned  zero_blocks  = (unsigned)((acc_f4 + 255) / 256);
    const unsigned  node_blocks  = (unsigned)((n_nodes + 31) / 32);

    float* hrcur = hra;
    float* hrnxt = hrb;

    for (int l = 0; l < 3; ++l) {
        const int relu = (l < 2) ? 1 : 0;

        zero_kernel<<<zero_blocks, 256, 0, stream>>>((float4*)acc, acc_f4);

        edge_kernel<<<edge_blocks, 256, 0, stream>>>(hin[l], hrcur, src, dst, etype, edir,
                                                     acc, edge_total16, n_nodes);

        node_kernel<<<node_blocks, 256, 0, stream>>>(hin[l], acc, hrcur + 2 * D,
                                                     W_S + (size_t)l * D * D,
                                                     W_O + (size_t)l * D * D,
                                                     W_I + (size_t)l * D * D,
                                                     hout[l], n_nodes, relu);

        if (l < 2) {
            rel_kernel<<<1, 192, 0, stream>>>(hrcur, W_rel + (size_t)l * D * D, hrnxt, 1);
            float* tmp = hrcur; hrcur = hrnxt; hrnxt = tmp;
        }
    }
}